// DilationLayerSum_37615323578483
// MI455X (gfx1250) — compile-verified
//
#include <hip/hip_runtime.h>
#include <stdint.h>

// out[b,i',i] = max_{dj',dj in [0,7)} ( x[b, i'+dj', i+dj] + w[dj',dj] )
// x: (B,13,13) f32, w: (7,7) f32, out: (B,7,7) f32

#define NB   64     // batches per block == threads per block (2 waves)
#define HH   13
#define WW   13
#define HW   169
#define KK   7
#define NOUT 49

typedef uint32_t v4u __attribute__((ext_vector_type(4)));
typedef int      v8i __attribute__((ext_vector_type(8)));
typedef int      v4i __attribute__((ext_vector_type(4)));

__global__ __launch_bounds__(NB)
void dilate7_kernel(const float* __restrict__ x, const float* __restrict__ wt,
                    float* __restrict__ out, long long nbatch) {
  __shared__ __align__(16) float lds[NB * HW];
  const int tid = threadIdx.x;
  const long long b0   = (long long)blockIdx.x * NB;
  const long long base = b0 * HW;                  // element offset of this block's tile
  const long long total = nbatch * (long long)HW;
  long long remll = total - base;
  const int tileElems = (int)(remll < (long long)(NB * HW) ? remll : (long long)(NB * HW));

  // ---------------- Stage tile -> LDS ----------------
#if __has_builtin(__builtin_amdgcn_tensor_load_to_lds) && __has_builtin(__builtin_amdgcn_s_wait_tensorcnt)
  if (tid < 32) {   // wave 0 issues one TDM DMA for the whole 43.3 KB tile
    uint64_t ga   = (uint64_t)(uintptr_t)(x + base);
    uint32_t loff = (uint32_t)(uintptr_t)(&lds[0]);   // flat LDS addr: low 32 bits = LDS byte offset
    uint32_t n    = (uint32_t)tileElems;              // elements (4B each), <= 10816 fits tile_dim0:16b
    // D# group 0: count=1 | lds_addr | global_addr[56:0] | type=2
    v4u g0;
    g0[0] = 1u;
    g0[1] = loff;
    g0[2] = (uint32_t)(ga & 0xFFFFFFFFull);
    g0[3] = (uint32_t)((ga >> 32) & 0x01FFFFFFull) | (2u << 30);
    // D# group 1: wg_mask=0, data_size=2 (4B), no barrier/iterate/pad,
    //             tensor_dim0=n, tensor_dim1=1, tile_dim0=n, tile_dim1=1, stride0=n
    v8i g1;
    g1[0] = (int)(2u << 16);
    g1[1] = (int)((n & 0xFFFFu) << 16);
    g1[2] = (int)(((n >> 16) & 0xFFFFu) | (1u << 16));
    g1[3] = (int)((n & 0xFFFFu) << 16);
    g1[4] = 1;
    g1[5] = (int)n;
    g1[6] = 0;
    g1[7] = 0;
    v4i g2; g2[0] = 1; g2[1] = 1; g2[2] = 0; g2[3] = 0;   // tensor_dim2=1, tensor_dim3=1
    v4i g3; g3[0] = 0; g3[1] = (int)(1u << 16); g3[2] = 0; g3[3] = 0; // tensor_dim4=1
#if defined(__clang_major__) && (__clang_major__ >= 23)
    // amdgpu-toolchain (clang-23): 6-arg form, trailing zero word-group
    // (g0.count==1 -> only the first descriptor is consumed).
    v8i g4; g4[0] = 0; g4[1] = 0; g4[2] = 0; g4[3] = 0;
            g4[4] = 0; g4[5] = 0; g4[6] = 0; g4[7] = 0;
    __builtin_amdgcn_tensor_load_to_lds(g0, g1, g2, g3, g4, 0);
#else
    // ROCm 7.2 (clang-22): 5-arg form
    __builtin_amdgcn_tensor_load_to_lds(g0, g1, g2, g3, 0);
#endif
    __builtin_amdgcn_s_wait_tensorcnt(0);
  }
#else
  {  // cooperative coalesced copy fallback (also what the host pass parses)
    const int nv = tileElems >> 2;
    const float4* __restrict__ src4 = (const float4*)(x + base);
    float4* dst4 = (float4*)lds;
    for (int i = tid; i < nv; i += NB) dst4[i] = src4[i];
    for (int i = (nv << 2) + tid; i < tileElems; i += NB) lds[i] = x[base + i];
  }
#endif
  __syncthreads();

  // ---------------- Weights: uniform -> SGPRs ----------------
  float w[NOUT];
#pragma unroll
  for (int k = 0; k < NOUT; ++k) w[k] = wt[k];

  // ---------------- Compute: thread-per-batch from LDS ----------------
  const long long b = b0 + tid;
  float* mine = &lds[tid * HW];   // lane stride 169 (mod 64 == 41): conflict-free
  if (b < nbatch) {
#pragma unroll 1
    for (int ip = 0; ip < KK; ++ip) {           // output row (kept rolled: ~6 KB body)
      float acc[KK];
#pragma unroll
      for (int i = 0; i < KK; ++i) acc[i] = -3.402823466e38f;
#pragma unroll
      for (int djp = 0; djp < KK; ++djp) {      // weight row -> constant indices
        float row[WW];
#pragma unroll
        for (int j = 0; j < WW; ++j) row[j] = mine[(ip + djp) * WW + j];
#pragma unroll
        for (int i = 0; i < KK; ++i) {
#pragma unroll
          for (int dj = 0; dj < KK; ++dj)
            acc[i] = fmaxf(acc[i], row[i + dj] + w[djp * KK + dj]);
        }
      }
      // park outputs in own LDS slot; indices 7*ip..7*ip+6 only touch rows <= ip,
      // which are never read by later ip iterations (7ip+6 < 13(ip+1)).
#pragma unroll
      for (int i = 0; i < KK; ++i) mine[ip * KK + i] = acc[i];
    }
  }
  __syncthreads();

  // ---------------- Coalesced write-back ----------------
  const long long ob = b0 * (long long)NOUT;
  long long vb = nbatch - b0;
  const int validB = (int)(vb < (long long)NB ? vb : (long long)NB);
  for (int idx = tid; idx < validB * NOUT; idx += NB) {
    const int bl = idx / NOUT;
    const int r  = idx - bl * NOUT;
    out[ob + idx] = lds[bl * HW + r];
  }
}

extern "C" void kernel_launch(void* const* d_in, const int* in_sizes, int n_in,
                              void* d_out, int out_size, void* d_ws, size_t ws_size,
                              hipStream_t stream) {
  const float* x  = (const float*)d_in[0];   // (B,13,13) f32
  const float* wt = (const float*)d_in[1];   // (7,7) f32
  float* out = (float*)d_out;                // (B,7,7) f32
  const long long nbatch = (long long)in_sizes[0] / HW;
  const int nblocks = (int)((nbatch + NB - 1) / NB);
  dilate7_kernel<<<nblocks, NB, 0, stream>>>(x, wt, out, nbatch);
}